// FSI_Initial_52862457479246
// MI455X (gfx1250) — compile-verified
//
#include <hip/hip_runtime.h>
#include <stdint.h>
#include <math.h>

// Problem geometry (fixed by setup_inputs)
#define B_    32
#define H_    512
#define W_    512
#define C_    4
#define HW    (H_ * W_)        // 262144
#define M_    (HW * C_)        // 1048576 elements per sample
#define M4    (M_ / 4)         // 262144 float4 per sample (== HW, one pixel's 4 channels)
#define HW4   (HW / 4)         // 65536
#define RSEL  174763           // ceil(M_ / 6)
#define NBIN  4096
#define CAP   16384            // candidate capacity per sample (expected ~256)

// Workspace layout (bytes)
#define WS_GHIST 0u            // B * NBIN * 4 = 524288
#define WS_BINFO 524288u       // B * 2 u32
#define WS_CNT   524800u       // B u32
#define WS_THR   525056u       // B f32
#define WS_INVN  525312u       // B f32
#define WS_ZPART 525568u       // B * 64 f32
#define WS_CAND  1048576u      // B * CAP f32 = 2 MB
#define WS_ZERO_WORDS (525568u / 4u)

// ---------------- CDNA5 async global->LDS helpers (ASYNCcnt path) ----------------
__device__ __forceinline__ void async_ld_b128(unsigned lds_byte_off, const void* gaddr) {
  // VDST = LDS byte address, VADDR = 64-bit global address (GV mode)
  asm volatile("global_load_async_to_lds_b128 %0, %1, off"
               :: "v"(lds_byte_off), "v"((unsigned long long)(uintptr_t)gaddr)
               : "memory");
}
__device__ __forceinline__ void wait_async_le1() { asm volatile("s_wait_asynccnt 0x1" ::: "memory"); }
__device__ __forceinline__ void wait_async_le0() { asm volatile("s_wait_asynccnt 0x0" ::: "memory"); }

// Monotonic value bin for Y in [0,1)
__device__ __forceinline__ unsigned ybin(float y) {
  int v = (int)(y * 4096.0f);
  v = v < 0 ? 0 : (v > 4095 ? 4095 : v);
  return (unsigned)v;
}

// ---------------- kernels ----------------
__global__ void k_zero(unsigned* __restrict__ w, unsigned nwords) {
  unsigned i = blockIdx.x * 256u + threadIdx.x;
  if (i < nwords) w[i] = 0u;
}

// Per-(sample, block) LDS histogram with double-buffered async staging.
// grid = (32, B), 256 threads, 32 iterations of one float4 per thread.
__global__ __launch_bounds__(256) void k_hist(const float* __restrict__ Y,
                                              unsigned* __restrict__ ghist) {
  __shared__ unsigned hist[NBIN];      // 16 KB
  __shared__ float4   stage[2][256];   // 2 x 4 KB staging
  const int tid = threadIdx.x;
  const int b   = blockIdx.y;

  for (int i = tid; i < NBIN; i += 256) hist[i] = 0u;
  __syncthreads();

  const float4* Y4 = (const float4*)Y;
  const float4* gp = Y4 + (size_t)b * M4 + (size_t)blockIdx.x * (256 * 32) + tid;
  const unsigned l0 = (unsigned)(uintptr_t)&stage[0][tid];
  const unsigned l1 = (unsigned)(uintptr_t)&stage[1][tid];

  async_ld_b128(l0, gp);               // prologue: tile 0 in flight
  #pragma unroll 1
  for (int it = 0; it < 32; ++it) {
    if (it + 1 < 32) {
      async_ld_b128((it & 1) ? l0 : l1, gp + (size_t)(it + 1) * 256);
      wait_async_le1();                // tile `it` complete (async loads retire in order)
    } else {
      wait_async_le0();
    }
    float4 v = stage[it & 1][tid];     // each lane reads only its own slot
    atomicAdd(&hist[ybin(v.x)], 1u);
    atomicAdd(&hist[ybin(v.y)], 1u);
    atomicAdd(&hist[ybin(v.z)], 1u);
    atomicAdd(&hist[ybin(v.w)], 1u);
  }
  __syncthreads();

  unsigned* gh = ghist + (size_t)b * NBIN;
  for (int i = tid; i < NBIN; i += 256) {
    unsigned c = hist[i];
    if (c) atomicAdd(&gh[i], c);
  }
}

// Descending scan: find bin holding the R-th largest, and residual rank within it.
__global__ void k_scan1(const unsigned* __restrict__ ghist, unsigned* __restrict__ binfo) {
  int b = blockIdx.x * blockDim.x + threadIdx.x;
  if (b >= B_) return;
  const unsigned* h = ghist + (size_t)b * NBIN;
  unsigned cum = 0, bin = 0, rp = 1;
  for (int i = NBIN - 1; i >= 0; --i) {
    unsigned c = h[i];
    if (cum + c >= (unsigned)RSEL) { bin = (unsigned)i; rp = (unsigned)RSEL - cum; break; }
    cum += c;
  }
  binfo[2 * b]     = bin;
  binfo[2 * b + 1] = rp;
}

// Gather candidates whose bin == b*.  grid = (32, B), 256 threads.
__global__ __launch_bounds__(256) void k_compact(const float* __restrict__ Y,
                                                 const unsigned* __restrict__ binfo,
                                                 unsigned* __restrict__ cnt,
                                                 float* __restrict__ cand) {
  const int b = blockIdx.y;
  const unsigned bstar = binfo[2 * b];
  const float4* Y4 = (const float4*)Y;
  size_t base = (size_t)b * M4 + (size_t)blockIdx.x * (256 * 32) + threadIdx.x;
  #pragma unroll 1
  for (int it = 0; it < 32; ++it) {
    float4 v = Y4[base + (size_t)it * 256];
    float a[4] = {v.x, v.y, v.z, v.w};
    #pragma unroll
    for (int k = 0; k < 4; ++k) {
      if (ybin(a[k]) == bstar) {
        unsigned idx = atomicAdd(&cnt[b], 1u);
        if (idx < CAP) cand[(size_t)b * CAP + idx] = a[k];
      }
    }
  }
}

// Exact r'-th largest among candidates via MSB radix-1 select on float bits
// (all values >= 0, so uint bit order == float order).  One block per sample.
__global__ __launch_bounds__(256) void k_select(const float* __restrict__ cand,
                                                const unsigned* __restrict__ cnt,
                                                const unsigned* __restrict__ binfo,
                                                float* __restrict__ thr) {
  __shared__ unsigned vals[CAP];       // 64 KB
  __shared__ unsigned red[256];
  __shared__ unsigned s_prefix, s_r;
  const int b = blockIdx.x;
  const int tid = threadIdx.x;

  unsigned n = cnt[b]; if (n > CAP) n = CAP;
  unsigned r = binfo[2 * b + 1];
  if (r < 1) r = 1;
  if (r > n) r = n;
  for (unsigned i = tid; i < n; i += 256)
    vals[i] = __float_as_uint(cand[(size_t)b * CAP + i]);
  if (tid == 0) { s_prefix = 0u; s_r = r; }
  __syncthreads();

  for (int bit = 31; bit >= 0; --bit) {
    unsigned pfx = s_prefix;
    unsigned hi  = (bit < 31) ? (~0u << (bit + 1)) : 0u;
    unsigned c = 0;
    for (unsigned i = tid; i < n; i += 256) {
      unsigned v = vals[i];
      if ((v & hi) == pfx && ((v >> bit) & 1u)) c++;
    }
    red[tid] = c; __syncthreads();
    for (int off = 128; off; off >>= 1) {
      if (tid < off) red[tid] += red[tid + off];
      __syncthreads();
    }
    if (tid == 0) {
      unsigned c1 = red[0];
      if (c1 >= s_r) s_prefix |= (1u << bit); else s_r -= c1;
    }
    __syncthreads();
  }
  if (tid == 0) thr[b] = __uint_as_float(s_prefix);   // threshold in Y-domain
}

// Mask + NHWC->NCHW transpose.  grid = (HW/256, B).
// (Y >= thr_Y) <=> (Y/512 >= thr_Y/512): scaling by 2^-9 is exact.
__global__ __launch_bounds__(256) void k_mask(const float* __restrict__ Y,
                                              const float* __restrict__ thr,
                                              float* __restrict__ out) {
  const int b = blockIdx.y;
  const int p = blockIdx.x * 256 + threadIdx.x;     // pixel index
  const float t = thr[b];
  const float4 v = ((const float4*)Y)[(size_t)b * HW + p];
  float* ob = out + (size_t)b * C_ * HW + p;
  ob[0 * (size_t)HW] = (v.x >= t) ? 1.0f : 0.0f;
  ob[1 * (size_t)HW] = (v.y >= t) ? 1.0f : 0.0f;
  ob[2 * (size_t)HW] = (v.z >= t) ? 1.0f : 0.0f;
  ob[3 * (size_t)HW] = (v.w >= t) ? 1.0f : 0.0f;
}

// Deterministic two-stage sum of |Z|^2.  grid = (64, B), 256 threads.
__global__ __launch_bounds__(256) void k_zsum(const float* __restrict__ Zabs,
                                              float* __restrict__ zpart) {
  __shared__ float red[256];
  const int b = blockIdx.y;
  const float4* A4 = (const float4*)Zabs;
  size_t base = (size_t)b * HW4 + (size_t)blockIdx.x * 1024 + threadIdx.x;
  float s = 0.0f;
  #pragma unroll
  for (int k = 0; k < 4; ++k) {
    float4 v = A4[base + (size_t)k * 256];
    s += v.x * v.x + v.y * v.y + v.z * v.z + v.w * v.w;
  }
  red[threadIdx.x] = s; __syncthreads();
  for (int off = 128; off; off >>= 1) {
    if (threadIdx.x < off) red[threadIdx.x] += red[threadIdx.x + off];
    __syncthreads();
  }
  if (threadIdx.x == 0) zpart[(size_t)b * 64 + blockIdx.x] = red[0];
}

__global__ __launch_bounds__(64) void k_zfinal(const float* __restrict__ zpart,
                                               float* __restrict__ invn) {
  __shared__ float red[64];
  const int b = blockIdx.x;
  red[threadIdx.x] = zpart[(size_t)b * 64 + threadIdx.x];
  __syncthreads();
  for (int off = 32; off; off >>= 1) {
    if (threadIdx.x < off) red[threadIdx.x] += red[threadIdx.x + off];
    __syncthreads();
  }
  if (threadIdx.x == 0) invn[b] = 1.0f / sqrtf(red[0]);
}

// Z0 = (|Z|/norm) * (cos a + i sin a), interleaved complex64.  grid = (HW/256, B).
__global__ __launch_bounds__(256) void k_zwrite(const float* __restrict__ Zabs,
                                                const float* __restrict__ Zang,
                                                const float* __restrict__ invn,
                                                float* __restrict__ out2) {
  const int b = blockIdx.y;
  const int p = blockIdx.x * 256 + threadIdx.x;
  const size_t i = (size_t)b * HW + p;
  float a = Zabs[i] * invn[b];
  float t = Zang[i];
  float s = sinf(t), c = cosf(t);
  ((float2*)out2)[i] = make_float2(a * c, a * s);
}

// ---------------- launch ----------------
extern "C" void kernel_launch(void* const* d_in, const int* in_sizes, int n_in,
                              void* d_out, int out_size, void* d_ws, size_t ws_size,
                              hipStream_t stream) {
  (void)in_sizes; (void)n_in; (void)out_size; (void)ws_size;
  const float* Y    = (const float*)d_in[0];
  const float* Zabs = (const float*)d_in[1];
  const float* Zang = (const float*)d_in[2];

  char* ws = (char*)d_ws;
  unsigned* ghist = (unsigned*)(ws + WS_GHIST);
  unsigned* binfo = (unsigned*)(ws + WS_BINFO);
  unsigned* cnt   = (unsigned*)(ws + WS_CNT);
  float*    thr   = (float*)(ws + WS_THR);
  float*    invn  = (float*)(ws + WS_INVN);
  float*    zpart = (float*)(ws + WS_ZPART);
  float*    cand  = (float*)(ws + WS_CAND);

  float* out_mask = (float*)d_out;
  float* out_z    = out_mask + (size_t)B_ * C_ * HW;   // 33,554,432 floats in

  hipLaunchKernelGGL(k_zero,   dim3((WS_ZERO_WORDS + 255) / 256), dim3(256), 0, stream,
                     (unsigned*)ws, WS_ZERO_WORDS);
  hipLaunchKernelGGL(k_hist,   dim3(32, B_),       dim3(256), 0, stream, Y, ghist);
  hipLaunchKernelGGL(k_scan1,  dim3(1),            dim3(32),  0, stream, ghist, binfo);
  hipLaunchKernelGGL(k_compact,dim3(32, B_),       dim3(256), 0, stream, Y, binfo, cnt, cand);
  hipLaunchKernelGGL(k_select, dim3(B_),           dim3(256), 0, stream, cand, cnt, binfo, thr);
  hipLaunchKernelGGL(k_mask,   dim3(HW / 256, B_), dim3(256), 0, stream, Y, thr, out_mask);
  hipLaunchKernelGGL(k_zsum,   dim3(64, B_),       dim3(256), 0, stream, Zabs, zpart);
  hipLaunchKernelGGL(k_zfinal, dim3(B_),           dim3(64),  0, stream, zpart, invn);
  hipLaunchKernelGGL(k_zwrite, dim3(HW / 256, B_), dim3(256), 0, stream, Zabs, Zang, invn, out_z);
}